// GRUDecoder_76570676953749
// MI455X (gfx1250) — compile-verified
//
#include <hip/hip_runtime.h>
#include <cmath>

#define VSZ   50257
#define EMBSZ 512
#define HDIM  512
#define BATCH 64
#define LSEQ  512
#define TSTEPS 32
#define XDIM  (EMBSZ + HDIM)   // 1024

typedef __attribute__((ext_vector_type(2))) float v2f;
typedef __attribute__((ext_vector_type(8))) float v8f;

// ---------------------------------------------------------------------------
// 16x16 f32 WMMA tile with compile-time K:  C = sum_k A(16xK) * B(Kx16)
//   A row-major, lda;  B(k,n) = W[n*ldw + k]
// Fragment layout per CDNA5 ISA (V_WMMA_F32_16X16X4_F32):
//   A: lane holds M=lane&15; vgpr0/1 hold K = k+2*(lane>>4), +1
//   B: lane holds N=lane&15; same K split across lane halves
// ---------------------------------------------------------------------------
template <int K>
__device__ __forceinline__ v8f wmma_tile_f32(const float* __restrict__ A, int lda,
                                             const float* __restrict__ W, int ldw,
                                             int lane)
{
    v8f c = {};
    const int half = lane >> 4;
    const int mn   = lane & 15;
    const float* arow = A + (size_t)mn * lda + 2 * half;
    const float* wrow = W + (size_t)mn * ldw + 2 * half;
#pragma unroll 8
    for (int k = 0; k < K; k += 4) {
        v2f a; a.x = arow[k]; a.y = arow[k + 1];
        v2f b; b.x = wrow[k]; b.y = wrow[k + 1];
        c = __builtin_amdgcn_wmma_f32_16x16x4_f32(false, a, false, b,
                                                  (short)0, c, false, false);
    }
    return c;
}

// ---------------------------------------------------------------------------
// ctx = (mean over valid l of encoded[b,l,:]) @ Wk.T   (exact: uniform softmax)
// ---------------------------------------------------------------------------
__global__ void enc_mean_kernel(const float* __restrict__ encoded,
                                const int* __restrict__ elen,
                                float* __restrict__ s)
{
    const int b = blockIdx.x;
    const int len = elen[b];
    const float* base = encoded + (size_t)b * LSEQ * HDIM;
    for (int h = threadIdx.x; h < HDIM; h += 256) {
        float acc = 0.f;
        for (int l = 0; l < len; ++l) acc += base[(size_t)l * HDIM + h];
        s[b * HDIM + h] = acc / (float)len;
    }
}

__global__ void ctx_proj_kernel(const float* __restrict__ s,
                                const float* __restrict__ Wk,
                                float* __restrict__ ctx)
{
    const int b = blockIdx.x;
    const float* sb = s + b * HDIM;
    for (int p = threadIdx.x; p < HDIM; p += 256) {
        const float* wrow = Wk + (size_t)p * HDIM;
        float acc = 0.f;
        for (int h = 0; h < HDIM; ++h) acc += sb[h] * wrow[h];
        ctx[b * HDIM + p] = acc;
    }
}

__global__ void init_kernel(float* __restrict__ h1, float* __restrict__ h2,
                            int* __restrict__ seq, const int* __restrict__ sos)
{
    const int i = blockIdx.x * blockDim.x + threadIdx.x;
    if (i < BATCH * HDIM) { h1[i] = 0.f; h2[i] = 0.f; }
    if (i < BATCH) seq[i] = sos[0];
}

__global__ void build_x_kernel(const float* __restrict__ emb,
                               const float* __restrict__ ctx,
                               const int* __restrict__ seq,
                               float* __restrict__ x)
{
    const int i = blockIdx.x * blockDim.x + threadIdx.x; // BATCH*XDIM
    const int b = i >> 10;
    const int k = i & (XDIM - 1);
    x[i] = (k < EMBSZ) ? emb[(size_t)seq[b] * EMBSZ + k]
                       : ctx[b * HDIM + (k - EMBSZ)];
}

// ---------------------------------------------------------------------------
// One GRU cell (torch GRUCell math), fully fused, K1 compile-time.
// Block i handles output columns [i*16, i*16+16).  8 waves:
//   wave (mt, grp): grp=0 -> gi tiles (A=X, K=K1), grp=1 -> gh tiles (A=Hprev)
//   each wave computes 3 gate tiles (r,z,n) via WMMA into LDS, then all
//   threads apply the exact gate nonlinearity.  grp is wave-uniform, so the
//   grp branch keeps EXEC all-ones inside each taken path.
// ---------------------------------------------------------------------------
template <int K1>
__global__ void gru_cell_kernel(const float* __restrict__ X,
                                const float* __restrict__ Hprev,
                                const float* __restrict__ Wih,
                                const float* __restrict__ Whh,
                                const float* __restrict__ bih,
                                const float* __restrict__ bhh,
                                float* __restrict__ Hout)
{
    __shared__ float G[2][3][BATCH][16];
    const int tid  = threadIdx.x;
    const int lane = tid & 31;
    const int w    = tid >> 5;
    const int mt   = w & 3;     // M-tile (16 rows of the 64-row batch)
    const int grp  = w >> 2;    // 0: input GEMM, 1: hidden GEMM (wave-uniform)
    const int i0   = blockIdx.x * 16;

    const int half = lane >> 4;
    const int nl   = lane & 15;
    if (grp == 0) {
        const float* Arow = X + (size_t)(mt * 16) * K1;
        for (int g = 0; g < 3; ++g) {
            v8f c = wmma_tile_f32<K1>(Arow, K1,
                                      Wih + (size_t)(g * HDIM + i0) * K1, K1, lane);
            for (int r = 0; r < 8; ++r)
                G[0][g][mt * 16 + r + 8 * half][nl] = c[r];
        }
    } else {
        const float* Arow = Hprev + (size_t)(mt * 16) * HDIM;
        for (int g = 0; g < 3; ++g) {
            v8f c = wmma_tile_f32<HDIM>(Arow, HDIM,
                                        Whh + (size_t)(g * HDIM + i0) * HDIM, HDIM, lane);
            for (int r = 0; r < 8; ++r)
                G[1][g][mt * 16 + r + 8 * half][nl] = c[r];
        }
    }
    __syncthreads();

    const int n   = tid & 15;
    const int col = i0 + n;
    const float br = bih[col],            bhr = bhh[col];
    const float bz = bih[HDIM + col],     bhz = bhh[HDIM + col];
    const float bn = bih[2 * HDIM + col], bhn = bhh[2 * HDIM + col];
    for (int mm = 0; mm < 4; ++mm) {
        const int m = (tid >> 4) + mm * 16;  // 0..63
        const float r  = 1.f / (1.f + expf(-(G[0][0][m][n] + br + G[1][0][m][n] + bhr)));
        const float z  = 1.f / (1.f + expf(-(G[0][1][m][n] + bz + G[1][1][m][n] + bhz)));
        const float nn = tanhf(G[0][2][m][n] + bn + r * (G[1][2][m][n] + bhn));
        const float hp = Hprev[(size_t)m * HDIM + col];
        Hout[(size_t)m * HDIM + col] = (1.f - z) * nn + z * hp;
    }
}

// ---------------------------------------------------------------------------
// logits = h2 @ Wout.T + bout  (64 x 50257, K=512), Wout L2-resident.
// Block: 16 batch rows x 512 vocab cols.  The 16x512 h2 tile is staged into
// LDS via gfx1250 async DMA (global_load_async_to_lds_b128 + ASYNCcnt).
// Each of the 8 waves owns 4 N-tiles: one A-fragment feeds 4 independent
// WMMA accumulator chains per k-step (ILP on the matrix pipe).
// ---------------------------------------------------------------------------
#define AS_LD 516   // padded LDS row stride (floats): conflict-free 16-lane reads
__global__ void logits_kernel(const float* __restrict__ H2,
                              const float* __restrict__ Wout,
                              const float* __restrict__ bout,
                              float* __restrict__ out)   // step-t slab (B x V)
{
    __shared__ float As[16 * AS_LD];
    const int tid = threadIdx.x;
    const int m0  = blockIdx.y * 16;

    // Async-stage A tile: 16 rows x 512 floats, 16B chunks (rows padded in LDS).
    for (int idx = tid; idx < 16 * (HDIM / 4); idx += 256) {
        const int mr = idx >> 7;            // 128 float4-chunks per row
        const int c4 = idx & 127;
        const unsigned lds_off =
            (unsigned)(uintptr_t)&As[mr * AS_LD + c4 * 4];  // low 32b = LDS offset
        const float* gp = H2 + (size_t)(m0 + mr) * HDIM + c4 * 4;
        asm volatile("global_load_async_to_lds_b128 %0, %1, off"
                     :: "v"(lds_off), "v"(gp) : "memory");
    }
    asm volatile("s_wait_asynccnt 0" ::: "memory");
    __syncthreads();

    const int lane  = tid & 31;
    const int w     = tid >> 5;
    const int nbase = blockIdx.x * 512 + w * 64;
    if (nbase >= VSZ) return;               // wave-uniform exit (EXEC stays full)

    const int half = lane >> 4;
    const int nl   = lane & 15;
    const float* arow = &As[nl * AS_LD + 2 * half];

    int ncol[4];
    const float* wrow[4];
#pragma unroll
    for (int j = 0; j < 4; ++j) {
        ncol[j] = nbase + j * 16 + nl;
        const int nc = ncol[j] < VSZ ? ncol[j] : VSZ - 1;   // clamp partial tile
        wrow[j] = Wout + (size_t)nc * HDIM + 2 * half;
    }

    v8f c0 = {}, c1 = {}, c2 = {}, c3 = {};
#pragma unroll 4
    for (int k = 0; k < HDIM; k += 4) {
        v2f a; a.x = arow[k]; a.y = arow[k + 1];
        v2f b0; b0.x = wrow[0][k]; b0.y = wrow[0][k + 1];
        v2f b1; b1.x = wrow[1][k]; b1.y = wrow[1][k + 1];
        v2f b2; b2.x = wrow[2][k]; b2.y = wrow[2][k + 1];
        v2f b3; b3.x = wrow[3][k]; b3.y = wrow[3][k + 1];
        c0 = __builtin_amdgcn_wmma_f32_16x16x4_f32(false, a, false, b0, (short)0, c0, false, false);
        c1 = __builtin_amdgcn_wmma_f32_16x16x4_f32(false, a, false, b1, (short)0, c1, false, false);
        c2 = __builtin_amdgcn_wmma_f32_16x16x4_f32(false, a, false, b2, (short)0, c2, false, false);
        c3 = __builtin_amdgcn_wmma_f32_16x16x4_f32(false, a, false, b3, (short)0, c3, false, false);
    }

    v8f acc[4] = {c0, c1, c2, c3};
#pragma unroll
    for (int j = 0; j < 4; ++j) {
        if (ncol[j] < VSZ) {
            const float bb = bout[ncol[j]];
            for (int r = 0; r < 8; ++r) {
                const int m = m0 + r + 8 * half;
                out[(size_t)m * VSZ + ncol[j]] = acc[j][r] + bb;
            }
        }
    }
}

// ---------------------------------------------------------------------------
// Greedy argmax per batch row (first-occurrence tie-break, like jnp.argmax)
// ---------------------------------------------------------------------------
__global__ void argmax_kernel(const float* __restrict__ logits, int* __restrict__ seq)
{
    __shared__ float sv[256];
    __shared__ int   si[256];
    const int b = blockIdx.x;
    const int tid = threadIdx.x;
    const float* row = logits + (size_t)b * VSZ;
    float bv = -INFINITY; int bi = VSZ;
    for (int v = tid; v < VSZ; v += 256) {
        const float x = row[v];
        if (x > bv) { bv = x; bi = v; }      // strided order is increasing v
    }
    sv[tid] = bv; si[tid] = bi;
    __syncthreads();
    for (int s = 128; s > 0; s >>= 1) {
        if (tid < s) {
            const float ov = sv[tid + s]; const int oi = si[tid + s];
            if (ov > sv[tid] || (ov == sv[tid] && oi < si[tid])) { sv[tid] = ov; si[tid] = oi; }
        }
        __syncthreads();
    }
    if (tid == 0) seq[b] = si[0];
}

// ---------------------------------------------------------------------------
extern "C" void kernel_launch(void* const* d_in, const int* in_sizes, int n_in,
                              void* d_out, int out_size, void* d_ws, size_t ws_size,
                              hipStream_t stream)
{
    const float* encoded = (const float*)d_in[0];
    const int*   enclen  = (const int*)  d_in[1];
    const int*   sos     = (const int*)  d_in[2];
    // d_in[3] = generated_seqlen (static 32 per reference shapes)
    const float* emb     = (const float*)d_in[4];
    const float* Wk      = (const float*)d_in[5];
    const float* Wih1    = (const float*)d_in[6];
    const float* Whh1    = (const float*)d_in[7];
    const float* bih1    = (const float*)d_in[8];
    const float* bhh1    = (const float*)d_in[9];
    const float* Wih2    = (const float*)d_in[10];
    const float* Whh2    = (const float*)d_in[11];
    const float* bih2    = (const float*)d_in[12];
    const float* bhh2    = (const float*)d_in[13];
    const float* Wout    = (const float*)d_in[14];
    const float* bout    = (const float*)d_in[15];
    float* outf = (float*)d_out;

    float* ws     = (float*)d_ws;
    float* s_mean = ws;                                   // 64*512
    float* ctx    = ws + 32768;                           // 64*512
    float* xbuf   = ws + 65536;                           // 64*1024
    float* h1a    = ws + 131072;
    float* h1b    = ws + 163840;
    float* h2a    = ws + 196608;
    float* h2b    = ws + 229376;
    int*   seq    = (int*)(ws + 262144);                  // 64 ints

    enc_mean_kernel<<<BATCH, 256, 0, stream>>>(encoded, enclen, s_mean);
    ctx_proj_kernel<<<BATCH, 256, 0, stream>>>(s_mean, Wk, ctx);
    init_kernel<<<(BATCH * HDIM + 255) / 256, 256, 0, stream>>>(h1a, h2a, seq, sos);

    float* h1c = h1a; float* h1n = h1b;
    float* h2c = h2a; float* h2n = h2b;
    for (int t = 0; t < TSTEPS; ++t) {
        build_x_kernel<<<(BATCH * XDIM + 255) / 256, 256, 0, stream>>>(emb, ctx, seq, xbuf);
        gru_cell_kernel<XDIM><<<HDIM / 16, 256, 0, stream>>>(xbuf, h1c,
                                                             Wih1, Whh1, bih1, bhh1, h1n);
        gru_cell_kernel<HDIM><<<HDIM / 16, 256, 0, stream>>>(h1n, h2c,
                                                             Wih2, Whh2, bih2, bhh2, h2n);
        float* outt = outf + (size_t)t * BATCH * VSZ;
        dim3 lgrid((VSZ + 511) / 512, BATCH / 16);
        logits_kernel<<<lgrid, 256, 0, stream>>>(h2n, Wout, bout, outt);
        argmax_kernel<<<BATCH, 256, 0, stream>>>(outt, seq);
        float* tmp;
        tmp = h1c; h1c = h1n; h1n = tmp;
        tmp = h2c; h2c = h2n; h2n = tmp;
    }
}